// SA_26431228740010
// MI455X (gfx1250) — compile-verified
//
#include <hip/hip_runtime.h>
#include <hip/hip_bf16.h>
#include <math.h>

// ---------------- problem constants ----------------
#define B_       8
#define N_       8192
#define NPOINT_  1024
#define NSAMPLE_ 32
#define RAD2_    0.04f          // 0.2^2
#define NCOL_    (B_ * NPOINT_ * NSAMPLE_)   // 262144 grouped columns
#define BN_EPS_  1e-5f

typedef float v2f __attribute__((ext_vector_type(2)));
typedef float v8f __attribute__((ext_vector_type(8)));

// =====================================================================
// 1) Farthest point sampling: one workgroup per batch, points + running
//    distance live in LDS; two-stage wave32 shuffle argmax per iteration.
// =====================================================================
__global__ void fps_kernel(const float* __restrict__ x, int* __restrict__ cents) {
    const int b = blockIdx.x;
    extern __shared__ float smem[];
    float* sx   = smem;                 // N_*3 floats
    float* dist = sx + N_ * 3;          // N_ floats
    float* wmax = dist + N_;            // 32 floats (per-wave partial)
    int*   widx = (int*)(wmax + 32);    // 32 ints
    __shared__ int s_far;

    const float* xb = x + (size_t)b * N_ * 3;
    const int t = threadIdx.x;          // 1024 threads
    for (int i = t; i < N_ * 3; i += 1024) sx[i] = xb[i];
    for (int i = t; i < N_;     i += 1024) dist[i] = 1e10f;
    if (t == 0) s_far = 0;
    __syncthreads();

    const int lane = t & 31, wid = t >> 5;
    for (int it = 0; it < NPOINT_; ++it) {
        const int far = s_far;
        if (t == 0) cents[b * NPOINT_ + it] = far;   // emit current centroid
        const float cx = sx[far * 3 + 0], cy = sx[far * 3 + 1], cz = sx[far * 3 + 2];

        float bestd = -1.0f; int besti = 0;
        for (int i = t; i < N_; i += 1024) {
            const float dx = sx[i * 3 + 0] - cx;
            const float dy = sx[i * 3 + 1] - cy;
            const float dz = sx[i * 3 + 2] - cz;
            float d = fminf(dist[i], dx * dx + dy * dy + dz * dz);
            dist[i] = d;
            if (d > bestd) { bestd = d; besti = i; }   // ascending scan => lowest idx on tie
        }
        // wave32 argmax (lowest index wins ties)
        #pragma unroll
        for (int off = 16; off > 0; off >>= 1) {
            float od = __shfl_down(bestd, off);
            int   oi = __shfl_down(besti, off);
            if (od > bestd || (od == bestd && oi < besti)) { bestd = od; besti = oi; }
        }
        if (lane == 0) { wmax[wid] = bestd; widx[wid] = besti; }
        __syncthreads();
        if (t < 32) {
            bestd = wmax[t]; besti = widx[t];
            #pragma unroll
            for (int off = 16; off > 0; off >>= 1) {
                float od = __shfl_down(bestd, off);
                int   oi = __shfl_down(besti, off);
                if (od > bestd || (od == bestd && oi < besti)) { bestd = od; besti = oi; }
            }
            if (t == 0) s_far = besti;
        }
        __syncthreads();
    }
}

// =====================================================================
// 2) Ball query + grouping. One block per batch (points in LDS), one
//    thread per centroid. Emits new_x and h0 in channel-major [4][NCOL]
//    layout (row 3 zero-padded so layer-1 K pads 3 -> 4 for WMMA).
// =====================================================================
__global__ void group_kernel(const float* __restrict__ x, const int* __restrict__ cents,
                             float* __restrict__ newx_out, float* __restrict__ h0) {
    const int b = blockIdx.x;
    extern __shared__ float sx[];       // N_*3 floats
    const float* xb = x + (size_t)b * N_ * 3;
    for (int i = threadIdx.x; i < N_ * 3; i += blockDim.x) sx[i] = xb[i];
    __syncthreads();

    const int s = threadIdx.x;          // 1024 centroids
    const int cent = cents[b * NPOINT_ + s];
    const float nx = sx[cent * 3 + 0], ny = sx[cent * 3 + 1], nz = sx[cent * 3 + 2];
    const size_t bs = (size_t)b * NPOINT_ + s;
    newx_out[bs * 3 + 0] = nx;
    newx_out[bs * 3 + 1] = ny;
    newx_out[bs * 3 + 2] = nz;

    const size_t col = bs * NSAMPLE_;
    int cnt = 0;
    float fx = 0.f, fy = 0.f, fz = 0.f;
    for (int i = 0; i < N_ && cnt < NSAMPLE_; ++i) {
        const float dx = sx[i * 3 + 0] - nx;
        const float dy = sx[i * 3 + 1] - ny;
        const float dz = sx[i * 3 + 2] - nz;
        if (dx * dx + dy * dy + dz * dz <= RAD2_) {
            if (cnt == 0) { fx = dx; fy = dy; fz = dz; }
            h0[0 * (size_t)NCOL_ + col + cnt] = dx;
            h0[1 * (size_t)NCOL_ + col + cnt] = dy;
            h0[2 * (size_t)NCOL_ + col + cnt] = dz;
            ++cnt;
        }
    }
    for (; cnt < NSAMPLE_; ++cnt) {     // pad with first in-ball point
        h0[0 * (size_t)NCOL_ + col + cnt] = fx;
        h0[1 * (size_t)NCOL_ + col + cnt] = fy;
        h0[2 * (size_t)NCOL_ + col + cnt] = fz;
    }
    for (int j = 0; j < NSAMPLE_; ++j)  // zero pad channel row 3 (WMMA K pad)
        h0[3 * (size_t)NCOL_ + col + j] = 0.f;
}

// =====================================================================
// 3) GEMM via V_WMMA_F32_16X16X4_F32.  One wave computes a 16x16 tile of
//    Y[O][NCOL] = W[O][KR] * H[K][NCOL] (K = KR rounded up to 4).
//    A frag (16x4 f32): lane m=lane&15; VGPR0/1 = K pair (lane>=16 -> K+2)
//    B frag (4x16 f32): lane n=lane&15; same K-pair split
//    C/D (16x16 f32):   VGPR r -> row r (lanes 0-15) / row r+8 (16-31)
// =====================================================================
template <int O, int K, int KR>
__global__ void gemm_wmma(const float* __restrict__ W, const float* __restrict__ H,
                          float* __restrict__ Y) {
    const int wave = (int)((blockIdx.x * blockDim.x + threadIdx.x) >> 5);
    const int lane = threadIdx.x & 31;
    constexpr int Mt = O / 16;
    const int m_tile = wave % Mt;
    const int n_tile = wave / Mt;

    const int half = lane >> 4;         // 0 or 1
    const int l15  = lane & 15;
    const int n    = n_tile * 16 + l15;
    const float* Wrow = W + (size_t)(m_tile * 16 + l15) * KR;

    v8f acc = {0.f, 0.f, 0.f, 0.f, 0.f, 0.f, 0.f, 0.f};
    #pragma unroll
    for (int k4 = 0; k4 < K; k4 += 4) {
        const int kb = k4 + (half << 1);
        v2f a, bfrag;
        if constexpr (KR % 4 != 0) {
            a.x = (kb + 0 < KR) ? Wrow[kb + 0] : 0.f;
            a.y = (kb + 1 < KR) ? Wrow[kb + 1] : 0.f;
        } else {
            a.x = Wrow[kb + 0];
            a.y = Wrow[kb + 1];
        }
        bfrag.x = H[(size_t)(kb + 0) * NCOL_ + n];
        bfrag.y = H[(size_t)(kb + 1) * NCOL_ + n];
        acc = __builtin_amdgcn_wmma_f32_16x16x4_f32(
            /*neg_a=*/false, a, /*neg_b=*/false, bfrag,
            /*c_mod=*/(short)0, acc, /*reuse_a=*/false, /*reuse_b=*/false);
    }
    float* Yb = Y + (size_t)(m_tile * 16 + half * 8) * NCOL_ + (size_t)n_tile * 16 + l15;
    #pragma unroll
    for (int r = 0; r < 8; ++r) Yb[(size_t)r * NCOL_] = acc[r];
}

// =====================================================================
// 4) Per-channel batch statistics: one block per channel, no atomics.
// =====================================================================
__global__ void bn_stats(const float* __restrict__ Y, float* __restrict__ mean,
                         float* __restrict__ rstd) {
    const int c = blockIdx.x;
    const float* row = Y + (size_t)c * NCOL_;
    float s = 0.f, s2 = 0.f;
    for (int i = threadIdx.x; i < NCOL_; i += blockDim.x) {
        const float v = row[i];
        s += v; s2 += v * v;
    }
    __shared__ float sh[256], sh2[256];
    sh[threadIdx.x] = s; sh2[threadIdx.x] = s2;
    __syncthreads();
    for (int off = 128; off > 0; off >>= 1) {
        if ((int)threadIdx.x < off) {
            sh[threadIdx.x]  += sh[threadIdx.x + off];
            sh2[threadIdx.x] += sh2[threadIdx.x + off];
        }
        __syncthreads();
    }
    if (threadIdx.x == 0) {
        const float m   = sh[0] * (1.0f / NCOL_);
        const float var = sh2[0] * (1.0f / NCOL_) - m * m;
        mean[c] = m;
        rstd[c] = rsqrtf(var + BN_EPS_);
    }
}

// =====================================================================
// 5) BN + ReLU (layers 1,2): elementwise, writes next-layer activations.
// =====================================================================
__global__ void bn_relu(const float* __restrict__ Y, float* __restrict__ Hout,
                        const float* __restrict__ mean, const float* __restrict__ rstd,
                        const float* __restrict__ g, const float* __restrict__ be, int O) {
    const size_t i = (size_t)blockIdx.x * blockDim.x + threadIdx.x;
    if (i >= (size_t)O * NCOL_) return;
    const int c = (int)(i / NCOL_);
    float v = g[c] * (Y[i] - mean[c]) * rstd[c] + be[c];
    Hout[i] = v > 0.f ? v : 0.f;
}

// =====================================================================
// 6) Layer-3 BN + ReLU fused with max over nsample; writes out[b,s,o].
// =====================================================================
__global__ void bn_relu_max(const float* __restrict__ Y3,
                            const float* __restrict__ mean, const float* __restrict__ rstd,
                            const float* __restrict__ g, const float* __restrict__ be,
                            float* __restrict__ out) {
    const int i = blockIdx.x * blockDim.x + threadIdx.x;    // B*NPOINT*128
    if (i >= B_ * NPOINT_ * 128) return;
    const int o  = i & 127;
    const int bs = i >> 7;
    const float* yrow = Y3 + (size_t)o * NCOL_ + (size_t)bs * NSAMPLE_;
    const float gm = g[o], bem = be[o], mn = mean[o], rs = rstd[o];
    float mx = 0.f;                      // post-ReLU values are >= 0
    #pragma unroll 8
    for (int j = 0; j < NSAMPLE_; ++j) {
        float v = gm * (yrow[j] - mn) * rs + bem;
        v = v > 0.f ? v : 0.f;
        mx = fmaxf(mx, v);
    }
    out[(size_t)bs * 128 + o] = mx;
}

// =====================================================================
// launch
// =====================================================================
extern "C" void kernel_launch(void* const* d_in, const int* in_sizes, int n_in,
                              void* d_out, int out_size, void* d_ws, size_t ws_size,
                              hipStream_t stream) {
    const float* x   = (const float*)d_in[0];
    const float* W1  = (const float*)d_in[1];   // (64,3)
    const float* g1  = (const float*)d_in[3];
    const float* be1 = (const float*)d_in[4];
    const float* W2  = (const float*)d_in[5];   // (64,64)
    const float* g2  = (const float*)d_in[7];
    const float* be2 = (const float*)d_in[8];
    const float* W3  = (const float*)d_in[9];   // (128,64)
    const float* g3  = (const float*)d_in[11];
    const float* be3 = (const float*)d_in[12];
    // (biases b1/b2/b3 cancel inside (y - mean) of BatchNorm; skipped)

    float* out_f   = (float*)d_out;
    float* newx    = out_f;                                   // B*1024*3
    float* out_feat= out_f + (size_t)B_ * NPOINT_ * 3;        // B*1024*128

    char* ws = (char*)d_ws;
    int*   cents = (int*)ws;                                  // 8192 ints
    float* mean  = (float*)(ws + 32768);                      // 128 floats
    float* rstd  = mean + 128;
    float* bufH  = (float*)(ws + 65536);                      // up to 128 x NCOL
    float* bufY  = bufH + (size_t)128 * NCOL_;                // up to 128 x NCOL

    // 1) FPS: 8 blocks x 1024 threads, 131328 B dynamic LDS
    const size_t fps_lds = (size_t)(N_ * 3 + N_ + 32) * 4 + 32 * 4;
    fps_kernel<<<B_, 1024, fps_lds, stream>>>(x, cents);

    // 2) ball query + grouping: 8 blocks x 1024 threads, 96 KB LDS
    group_kernel<<<B_, 1024, (size_t)N_ * 3 * 4, stream>>>(x, cents, newx, bufH);

    const int elemThreads = 256;
    // ---- layer 1: K=3 (padded 4), O=64 ----
    gemm_wmma<64, 4, 3><<<(64 / 16) * (NCOL_ / 16) / 8, 256, 0, stream>>>(W1, bufH, bufY);
    bn_stats<<<64, 256, 0, stream>>>(bufY, mean, rstd);
    {
        const size_t n = (size_t)64 * NCOL_;
        bn_relu<<<(int)((n + elemThreads - 1) / elemThreads), elemThreads, 0, stream>>>(
            bufY, bufH, mean, rstd, g1, be1, 64);
    }
    // ---- layer 2: K=64, O=64 ----
    gemm_wmma<64, 64, 64><<<(64 / 16) * (NCOL_ / 16) / 8, 256, 0, stream>>>(W2, bufH, bufY);
    bn_stats<<<64, 256, 0, stream>>>(bufY, mean, rstd);
    {
        const size_t n = (size_t)64 * NCOL_;
        bn_relu<<<(int)((n + elemThreads - 1) / elemThreads), elemThreads, 0, stream>>>(
            bufY, bufH, mean, rstd, g2, be2, 64);
    }
    // ---- layer 3: K=64, O=128, fused BN+ReLU+maxpool ----
    gemm_wmma<128, 64, 64><<<(128 / 16) * (NCOL_ / 16) / 8, 256, 0, stream>>>(W3, bufH, bufY);
    bn_stats<<<128, 256, 0, stream>>>(bufY, mean, rstd);
    {
        const int n = B_ * NPOINT_ * 128;
        bn_relu_max<<<(n + elemThreads - 1) / elemThreads, elemThreads, 0, stream>>>(
            bufY, mean, rstd, g3, be3, out_feat);
    }
}